// QueryReference_12257836663096
// MI455X (gfx1250) — compile-verified
//
#include <hip/hip_runtime.h>
#include <stdint.h>

#define BLOCK   256
#define NSET    9     // 1 query + 8 refs
#define NREF    8
#define NDIM    16
#define NGROUP  4

#define AS1 __attribute__((address_space(1)))
#define AS3 __attribute__((address_space(3)))

typedef int v4i __attribute__((ext_vector_type(4)));

// ---- CDNA5 async global->LDS copy (ASYNCcnt path, CDNA5 ISA) ----
__device__ __forceinline__ void async_g2l_b128(const void* g, void* l) {
#if __has_builtin(__builtin_amdgcn_global_load_async_to_lds_b128)
    __builtin_amdgcn_global_load_async_to_lds_b128(
        (AS1 v4i*)(uintptr_t)g,
        (AS3 v4i*)(uint32_t)(uintptr_t)l,
        0, 0);
#else
    uint32_t lo = (uint32_t)(uintptr_t)l;
    uint64_t ga = (uint64_t)(uintptr_t)g;
    asm volatile("global_load_async_to_lds_b128 %0, %1, off" :: "v"(lo), "v"(ga) : "memory");
#endif
}

__device__ __forceinline__ void wait_async0() {
#if __has_builtin(__builtin_amdgcn_s_wait_asynccnt)
    __builtin_amdgcn_s_wait_asynccnt(0);
#else
    asm volatile("s_wait_asynccnt 0" ::: "memory");
#endif
}

// bytes % 16 == 0, 16B-aligned src & dst; all BLOCK threads participate
__device__ __forceinline__ void block_copy_async(const char* __restrict__ g, char* l, int bytes) {
    for (int o = threadIdx.x * 16; o < bytes; o += BLOCK * 16)
        async_g2l_b128(g + o, l + o);
}

// tail fallback: bytes % 4 == 0
__device__ __forceinline__ void block_copy_sync(const char* __restrict__ g, char* l, int bytes) {
    for (int o = threadIdx.x * 4; o < bytes; o += BLOCK * 4)
        *(int*)(l + o) = *(const int*)(g + o);
}

__device__ __forceinline__ float wdist2(float4 w, float4 q, float4 a) {
    float dx = q.x - a.x, dy = q.y - a.y, dz = q.z - a.z, dw = q.w - a.w;
    return w.x * dx * dx + w.y * dy * dy + w.z * dz * dz + w.w * dw * dw;
}

__global__ void __launch_bounds__(BLOCK)
simjudge_kernel(const int* __restrict__ stim,     // [n, 9]
                const int* __restrict__ cfg,      // [n]
                const int* __restrict__ grp,      // [n]
                const int* __restrict__ pres,     // [n, 9]
                const float* __restrict__ z,      // [10000, 16]
                const float* __restrict__ w,      // [4, 16]
                float* __restrict__ out,          // [n]
                int n)
{
    __shared__ __align__(16) int   s_stim[BLOCK * NSET];
    __shared__ __align__(16) int   s_pres[BLOCK * NSET];
    __shared__ __align__(16) int   s_cfg[BLOCK];
    __shared__ __align__(16) int   s_grp[BLOCK];
    __shared__ __align__(16) float s_w[NGROUP * NDIM];

    const int base  = blockIdx.x * BLOCK;
    const int count = min(BLOCK, n - base);

    if (count == BLOCK) {
        // fully coalesced async DMA into LDS; tracked by ASYNCcnt
        block_copy_async((const char*)(stim + (size_t)base * NSET), (char*)s_stim, BLOCK * NSET * 4);
        block_copy_async((const char*)(pres + (size_t)base * NSET), (char*)s_pres, BLOCK * NSET * 4);
        block_copy_async((const char*)(cfg  + base),               (char*)s_cfg,  BLOCK * 4);
        block_copy_async((const char*)(grp  + base),               (char*)s_grp,  BLOCK * 4);
        block_copy_async((const char*)w,                           (char*)s_w,    NGROUP * NDIM * 4);
        wait_async0();
    } else {
        block_copy_sync((const char*)(stim + (size_t)base * NSET), (char*)s_stim, count * NSET * 4);
        block_copy_sync((const char*)(pres + (size_t)base * NSET), (char*)s_pres, count * NSET * 4);
        block_copy_sync((const char*)(cfg  + base),               (char*)s_cfg,  count * 4);
        block_copy_sync((const char*)(grp  + base),               (char*)s_grp,  count * 4);
        block_copy_sync((const char*)w,                           (char*)s_w,    NGROUP * NDIM * 4);
    }
    __syncthreads();

    const int t     = threadIdx.x;
    const int trial = base + t;
    if (trial >= n) return;

    const int* ts = s_stim + t * NSET;
    const int* tp = s_pres + t * NSET;
    const int  g  = s_grp[t] & (NGROUP - 1);
    const int  c  = s_cfg[t] & 3;
    const int  nsel = (0x1321 >> (4 * c)) & 0xF;   // N_SELECT = {1,2,3,1}

    const float4* wr = (const float4*)(s_w + g * NDIM);
    const float4 w0 = wr[0], w1 = wr[1], w2 = wr[2], w3 = wr[3];

    const float4* zq = (const float4*)(z + (size_t)ts[0] * NDIM);   // 64B-aligned row, L2-resident
    const float4 q0 = zq[0], q1 = zq[1], q2 = zq[2], q3 = zq[3];

    float sim[NREF];
#pragma unroll
    for (int r = 0; r < NREF; ++r) {
        const float4* zr = (const float4*)(z + (size_t)ts[1 + r] * NDIM);
        float4 a0 = zr[0], a1 = zr[1], a2 = zr[2], a3 = zr[3];
        float d2 = wdist2(w0, q0, a0) + wdist2(w1, q1, a1)
                 + wdist2(w2, q2, a2) + wdist2(w3, q3, a3);
        float d  = sqrtf(d2);                       // v_sqrt_f32
        float s  = __expf(-d) + 0.001f;             // v_exp_f32 (BETA=1, GAMMA=1e-3)
        sim[r] = s * (float)tp[1 + r];
    }

    // suffix[r] = sum_{j>=r} sim[j]
    float suf[NREF];
    float run = 0.0f;
#pragma unroll
    for (int r = NREF - 1; r >= 0; --r) { run += sim[r]; suf[r] = run; }

    float like = 1.0f;
#pragma unroll
    for (int r = 0; r < 3; ++r)                     // n_select <= 3
        if (r < nsel) like *= sim[r] / suf[r];

    out[trial] = like;
}

extern "C" void kernel_launch(void* const* d_in, const int* in_sizes, int n_in,
                              void* d_out, int out_size, void* d_ws, size_t ws_size,
                              hipStream_t stream) {
    const int*   stim = (const int*)d_in[0];
    const int*   cfg  = (const int*)d_in[1];
    const int*   grp  = (const int*)d_in[2];
    const int*   pres = (const int*)d_in[3];
    const float* z    = (const float*)d_in[4];
    const float* w    = (const float*)d_in[5];
    float*       out  = (float*)d_out;

    const int n = out_size;                 // N_TRIAL
    const int blocks = (n + BLOCK - 1) / BLOCK;
    simjudge_kernel<<<blocks, BLOCK, 0, stream>>>(stim, cfg, grp, pres, z, w, out, n);
}